// RBRLoss_68616397521110
// MI455X (gfx1250) — compile-verified
//
#include <hip/hip_runtime.h>
#include <hip/hip_bf16.h>

typedef __attribute__((ext_vector_type(2))) float v2f;
typedef __attribute__((ext_vector_type(4))) float v4f;
typedef __attribute__((ext_vector_type(8))) float v8f;
typedef __attribute__((ext_vector_type(4))) int   v4i;

#define DIM      128
#define SIGMA    0.5f
#define MAX_ITER 1000
#define LR       0.0316227766016838f   /* 1/sqrt(1000) */
#define ZETA     1e-6f
#define SQRT_P   11.31370849898476f    /* sqrt(128) */
#define PE_RAD   0.08838834764831845f  /* 1/sqrt(128) */
#define INV_PM1  (1.0f / 127.0f)

#if __has_builtin(__builtin_amdgcn_global_load_async_to_lds_b128)
#define HAVE_ASYNC_LDS 1
#else
#define HAVE_ASYNC_LDS 0
#endif

// exact pointee type required by the builtin: int4 in AS(1)/AS(3)
typedef __attribute__((address_space(1))) v4i* glb_v4i_ptr;
typedef __attribute__((address_space(3))) v4i* lds_v4i_ptr;

static __device__ __forceinline__ void wait_asynccnt0() {
#if __has_builtin(__builtin_amdgcn_s_wait_asynccnt)
    __builtin_amdgcn_s_wait_asynccnt(0);
#else
    asm volatile("s_wait_asynccnt 0" ::: "memory");
#endif
}

// ---------------------------------------------------------------------------
// Stage 1: c[i] = || x - X[i] ||  via f32 WMMA Gram-diagonal.
// One wave = 16 rows. X tile (16x128 f32, 8KB) is staged into LDS with
// GLOBAL_LOAD_ASYNC_TO_LDS_B128 (ASYNCcnt-tracked async copy engine), then
// the wave builds diff tiles and chains 32x V_WMMA_F32_16X16X4_F32 with
// A = B = diff (register-identical layouts) so the accumulator diagonal
// is sum_k diff^2 = c^2.
// ---------------------------------------------------------------------------
__global__ __launch_bounds__(32)
void rbr_dist_wmma(const float* __restrict__ x,
                   const float* __restrict__ X,
                   float* __restrict__ c,
                   int n)
{
    __shared__ float xs[DIM];            // broadcast copy of x
    __shared__ float xtile[16 * DIM];    // 16-row tile of X (8 KB)
    __shared__ float gram[32 * 8];       // accumulator spill for diag extract

    const int lane = threadIdx.x;        // 0..31, wave32
    #pragma unroll
    for (int j = 0; j < DIM / 32; ++j)
        xs[lane + 32 * j] = x[lane + 32 * j];

    const int m0 = blockIdx.x * 16;

    if (m0 + 16 <= n) {
#if HAVE_ASYNC_LDS
        // 16 wave-wide async B128 copies move the whole 8KB tile.
        const char* gbase = (const char*)(X + (size_t)m0 * DIM);
        char*       lbase = (char*)xtile;
        #pragma unroll
        for (int j = 0; j < 16; ++j) {
            const int off = j * 512 + lane * 16;
            __builtin_amdgcn_global_load_async_to_lds_b128(
                (glb_v4i_ptr)(gbase + off), (lds_v4i_ptr)(lbase + off), 0, 0);
        }
        wait_asynccnt0();
#else
        const v4f* gp = (const v4f*)(X + (size_t)m0 * DIM);
        v4f*       lp = (v4f*)xtile;
        #pragma unroll
        for (int j = 0; j < 16; ++j)
            lp[j * 32 + lane] = gp[j * 32 + lane];
#endif
    } else {
        // rare ragged tail: clamped scalar fill
        for (int j = 0; j < 16; ++j) {
            const int r = min(m0 + j, n - 1);
            #pragma unroll
            for (int t = 0; t < DIM / 32; ++t)
                xtile[j * DIM + lane + 32 * t] = X[(size_t)r * DIM + lane + 32 * t];
        }
    }
    __syncthreads();

    const int mrow = lane & 15;          // M index this lane carries
    const int ksub = (lane >> 4) * 2;    // K 0/1 (lanes 0-15) vs 2/3 (lanes 16-31)
    const float* rp = &xtile[mrow * DIM + ksub];
    const float* xp = &xs[ksub];

    v8f acc = {};
    #pragma unroll
    for (int k0 = 0; k0 < DIM; k0 += 4) {
        v2f dv;
        dv.x = xp[k0]     - rp[k0];
        dv.y = xp[k0 + 1] - rp[k0 + 1];
        // D = A*B + C with A = B = diff tile
        acc = __builtin_amdgcn_wmma_f32_16x16x4_f32(
                  false, dv, false, dv, (short)0, acc, false, false);
    }

    #pragma unroll
    for (int j = 0; j < 8; ++j)
        gram[lane * 8 + j] = acc[j];
    __syncthreads();

    if (lane < 16) {
        const int m   = lane;
        // D[m][m]: m<8 -> lane m, vgpr m ; m>=8 -> lane 16+m, vgpr m-8
        const int src = (m < 8) ? (m * 8 + m) : ((16 + m) * 8 + (m - 8));
        const float d2 = fmaxf(gram[src], 0.0f);
        const int r = m0 + m;
        if (r < n)
            c[r] = __builtin_amdgcn_sqrtf(d2);
    }
}

// ---------------------------------------------------------------------------
// Packed-f32 helpers (lower to V_PK_ADD/MUL/FMA_F32 on CDNA5)
// ---------------------------------------------------------------------------
static __device__ __forceinline__ v2f vmax0(v2f a) {
    a.x = fmaxf(a.x, 0.0f); a.y = fmaxf(a.y, 0.0f); return a;
}
static __device__ __forceinline__ v2f vrcp(v2f a) {
    v2f r; r.x = __builtin_amdgcn_rcpf(a.x); r.y = __builtin_amdgcn_rcpf(a.y); return r;
}
static __device__ __forceinline__ v2f vsqrt(v2f a) {
    v2f r; r.x = __builtin_amdgcn_sqrtf(a.x); r.y = __builtin_amdgcn_sqrtf(a.y); return r;
}
static __device__ __forceinline__ v2f vrsq(v2f a) {
    v2f r; r.x = __builtin_amdgcn_rsqf(a.x); r.y = __builtin_amdgcn_rsqf(a.y); return r;
}
static __device__ __forceinline__ v2f vmin1(v2f a) {
    a.x = fminf(a.x, 1.0f); a.y = fminf(a.y, 1.0f); return a;
}

// ---------------------------------------------------------------------------
// Stage 2a: optimistic PGD. Two rows per thread as v2f so the FMA-class math
// packs into V_PK_*_F32 (2x f32 rate); 1000 register-resident iterations.
// ---------------------------------------------------------------------------
__global__ void rbr_pgd_op(const float* __restrict__ c_in,
                           float* __restrict__ numer,
                           int n)
{
    const int i  = blockIdx.x * blockDim.x + threadIdx.x;
    const int r0 = 2 * i;
    if (r0 >= n) return;
    const int r1 = min(r0 + 1, n - 1);

    v2f cc; cc.x = c_in[r0]; cc.y = c_in[r1];
    v2f v0 = {0.0f, 0.0f}, v1 = {0.0f, 0.0f};

    for (int it = 0; it < MAX_ITER; ++it) {
        v2f d   = v1 + SIGMA;
        v2f r   = cc - v0;
        v2f id  = vrcp(d);
        v2f id2 = id * id;
        v2f g0  = -(r * id2);
        v2f g1  = id - r * r * id2 * id;
        v0 = vmax0(v0 - LR * g0);
        v1 = vmax0(v1 - LR * g1);
        v2f ns  = v0 * v0 + v1 * v1;
        v2f sc  = vmin1(vrsq(ns));           // radius = 1
        v0 *= sc; v1 *= sc;
    }

    v2f d = v1 + SIGMA;
    v2f r = cc - v0;
    const float cl = 127.0f * __logf(SIGMA);
    float Lx = __logf(d.x) + r.x * r.x / (2.0f * d.x * d.x) + cl;
    float Ly = __logf(d.y) + r.y * r.y / (2.0f * d.y * d.y) + cl;
    numer[r0] = Lx;
    if (r0 + 1 < n) numer[r0 + 1] = Ly;
}

// ---------------------------------------------------------------------------
// Stage 2b: pessimistic PGD (denom), packed 2 rows/thread.
// 1/d and 1/ff share one v_rcp via ip = rcp(d*ff): 1/d = ff*ip, 1/ff = d*ip.
// ---------------------------------------------------------------------------
__global__ void rbr_pgd_pe(const float* __restrict__ c_in,
                           float* __restrict__ denom,
                           int n)
{
    const int i  = blockIdx.x * blockDim.x + threadIdx.x;
    const int r0 = 2 * i;
    if (r0 >= n) return;
    const int r1 = min(r0 + 1, n - 1);

    v2f cc; cc.x = c_in[r0]; cc.y = c_in[r1];
    v2f u0 = {0.0f, 0.0f}, u1 = {0.0f, 0.0f};

    for (int it = 0; it < MAX_ITER; ++it) {
        v2f d      = u1 + SIGMA;
        v2f s      = cc + SQRT_P * u0;
        v2f inside = ((ZETA + 1.0f) - 128.0f * u0 * u0 - u1 * u1) * INV_PM1;
        v2f f      = vsqrt(inside);
        v2f ff     = f * (f + SIGMA);
        v2f ip     = vrcp(d * ff);
        v2f id     = ff * ip;                // 1/d
        v2f iff    = d * ip;                 // 1/ff
        v2f id2    = id * id;
        v2f g0     = -(SQRT_P * s * id2) - 128.0f * u0 * iff;
        v2f g1     = -id + s * s * id2 * id + u1 * iff;
        u0 = vmax0(u0 - LR * g0);
        u1 = vmax0(u1 - LR * g1);
        v2f ns     = u0 * u0 + u1 * u1;
        v2f sc     = vmin1(PE_RAD * vrsq(ns));
        u0 *= sc; u1 *= sc;
    }

    v2f d      = u1 + SIGMA;
    v2f s      = cc + SQRT_P * u0;
    v2f inside = ((ZETA + 1.0f) - 128.0f * u0 * u0 - u1 * u1) * INV_PM1;
    v2f f      = vsqrt(inside);
    float Lx = -__logf(d.x) - s.x * s.x / (2.0f * d.x * d.x) - 127.0f * __logf(f.x + SIGMA);
    float Ly = -__logf(d.y) - s.y * s.y / (2.0f * d.y * d.y) - 127.0f * __logf(f.y + SIGMA);
    denom[r0] = Lx;
    if (r0 + 1 < n) denom[r0 + 1] = Ly;
}

// ---------------------------------------------------------------------------
// Stage 3: F = numer - denom
// ---------------------------------------------------------------------------
__global__ void rbr_combine(const float* __restrict__ numer,
                            const float* __restrict__ denom,
                            float* __restrict__ F,
                            int n)
{
    int i = blockIdx.x * blockDim.x + threadIdx.x;
    if (i < n) F[i] = numer[i] - denom[i];
}

extern "C" void kernel_launch(void* const* d_in, const int* in_sizes, int n_in,
                              void* d_out, int out_size, void* d_ws, size_t ws_size,
                              hipStream_t stream) {
    const float* x    = (const float*)d_in[0];   // [128]
    // d_in[1] = X_feas: unused on the main path (both sets non-empty)
    const float* Xpos = (const float*)d_in[2];   // [n,128]
    const float* Xneg = (const float*)d_in[3];   // [n,128]

    const int d = in_sizes[0];          // 128
    const int n = in_sizes[2] / d;      // 100000
    (void)n_in; (void)out_size; (void)ws_size;

    float* out   = (float*)d_out;       // [F | denom | numer]
    float* Fout  = out;
    float* denom = out + n;
    float* numer = out + 2 * n;

    float* c_pos = (float*)d_ws;        // n floats
    float* c_neg = c_pos + n;           // n floats

    const int nwaves = (n + 15) / 16;   // one wave = 16 rows
    rbr_dist_wmma<<<nwaves, 32, 0, stream>>>(x, Xpos, c_pos, n);
    rbr_dist_wmma<<<nwaves, 32, 0, stream>>>(x, Xneg, c_neg, n);

    const int blk  = 256;               // 8 waves / workgroup on CDNA5
    const int half = (n + 1) / 2;       // 2 rows per thread (packed f32)
    const int nbh  = (half + blk - 1) / blk;
    rbr_pgd_op<<<nbh, blk, 0, stream>>>(c_pos, numer, n);
    rbr_pgd_pe<<<nbh, blk, 0, stream>>>(c_neg, denom, n);

    const int nb = (n + blk - 1) / blk;
    rbr_combine<<<nb, blk, 0, stream>>>(numer, denom, Fout, n);
}